// SparseAbacusLayer_77137612636835
// MI455X (gfx1250) — compile-verified
//
#include <hip/hip_runtime.h>
#include <hip/hip_bf16.h>

// SparseAbacusLayer: uniform-knot piecewise-linear interp + pairwise sum.
// Closed-form searchsorted (uniform knots) with branchless one-ULP boundary
// correction; activations (64MB) L2-resident; streamed buffers TH=NT.

#define N_IN_KNOTS 16777216

typedef float v4f __attribute__((ext_vector_type(4)));
typedef float v2f __attribute__((ext_vector_type(2)));

__device__ __forceinline__ float clamp01(float v) {
    return __builtin_fminf(__builtin_fmaxf(v, 0.0f), 1.0f);
}

// One interpolated sample; fully branchless, no short-circuit control flow.
__device__ __forceinline__ float interp1(const float* __restrict__ act, float x) {
    const float M    = (float)(N_IN_KNOTS - 1);          // 16777215.0f (exact)
    const float STEP = 1.0f / (float)(N_IN_KNOTS - 1);

    x = clamp01(x);

    // searchsorted(pos, x, 'left'): smallest i with pos[i] >= x.
    // Keep index in float domain: integer-valued, < 2^24, exact in f32.
    float fidx = __builtin_ceilf(x * M);                 // in [0, M]
    float pm1  = (fidx - 1.0f) * STEP;                   // pos[idx-1]
    float p0   = fidx * STEP;                            // pos[idx]

    // Mutually exclusive corrections (pm1 >= x implies p0 > x). Bitwise &
    // (not &&) so both compares evaluate unconditionally -> v_cmp/s_and/
    // v_cndmask, no exec-mask predication.
    int inc = (int)(p0 < x);
    int dec = (int)(pm1 >= x) & (int)(fidx > 0.5f);
    float adj = inc ? 1.0f : 0.0f;
    adj = dec ? -1.0f : adj;
    fidx = fidx + adj;
    fidx = __builtin_fminf(fidx, M);                     // x==1.0 safety clamp
    float pidx = fidx * STEP;

    // Slope between knots (j-1, j); for idx==0, offset==0 so slope is dead —
    // use j=1 to keep loads in-bounds, branchlessly.
    float fj = __builtin_fmaxf(fidx, 1.0f);
    int   j  = (int)fj;                                  // single cvt for addressing
    float y0 = clamp01(act[j - 1]);                      // adjacent b32 gathers,
    float y1 = clamp01(act[j]);                          // L2-hit (64MB << 192MB)
    float denom  = fj * STEP - (fj - 1.0f) * STEP;       // replicate float pos diff
    float offset = x - pidx;                             // <= 0; exactly 0 at idx==0
    float base   = (fidx > 0.5f) ? y1 : y0;              // y[idx]
    return __builtin_fmaf(__fdividef(y1 - y0, denom), offset, base);
}

// 2 outputs (4 samples) per thread: b128 NT load of coords, b64 NT store.
__global__ void __launch_bounds__(256)
sparse_abacus_interp_kernel(const float* __restrict__ act,
                            const v4f* __restrict__ sp4,
                            v2f* __restrict__ out2,
                            int n_pairs) {
    int i = blockIdx.x * blockDim.x + threadIdx.x;
    if (i >= n_pairs) return;

    v4f s = __builtin_nontemporal_load(&sp4[i]);   // streamed once: TH=NT

    v2f r;
    r.x = interp1(act, s.x) + interp1(act, s.y);
    r.y = interp1(act, s.z) + interp1(act, s.w);

    __builtin_nontemporal_store(r, &out2[i]);      // streamed once: TH=NT
}

// Scalar tail kernel (only used if out_size is odd; not the case here).
__global__ void sparse_abacus_interp_tail(const float* __restrict__ act,
                                          const float* __restrict__ sp,
                                          float* __restrict__ out,
                                          int start, int n_out) {
    int i = start + blockIdx.x * blockDim.x + threadIdx.x;
    if (i >= n_out) return;
    float sx = sp[2 * i];
    float sy = sp[2 * i + 1];
    out[i] = interp1(act, sx) + interp1(act, sy);
}

extern "C" void kernel_launch(void* const* d_in, const int* in_sizes, int n_in,
                              void* d_out, int out_size, void* d_ws, size_t ws_size,
                              hipStream_t stream) {
    (void)in_sizes; (void)n_in; (void)d_ws; (void)ws_size;

    const float* act = (const float*)d_in[0];   // activations, N_IN floats
    const float* sp  = (const float*)d_in[1];   // sample_points, out_size*2 floats
    float*       out = (float*)d_out;           // out_size floats

    int n_pairs = out_size >> 1;
    if (n_pairs > 0) {
        const int threads = 256;   // 8 wave32 waves per block
        int blocks = (n_pairs + threads - 1) / threads;
        sparse_abacus_interp_kernel<<<blocks, threads, 0, stream>>>(
            act, (const v4f*)sp, (v2f*)out, n_pairs);
    }
    if (out_size & 1) {
        sparse_abacus_interp_tail<<<1, 32, 0, stream>>>(
            act, sp, out, out_size - 1, out_size);
    }
}